// Angles2Backbone_35270271435034
// MI455X (gfx1250) — compile-verified
//
#include <hip/hip_runtime.h>

typedef float v2f __attribute__((ext_vector_type(2)));
typedef float v8f __attribute__((ext_vector_type(8)));

#define PI_F 3.14159265358979323846f

// Right-multiply rigid transform P (3x3 rotation cols + translation) by the
// analytic per-atom transform T(ca,sa,cb,sb,R):
//   T cols: (ca,sa,0) | (-sa*cb, ca*cb, sb) | (sa*sb, -ca*sb, cb) | (R*ca, R*sa, 0)
// P layout: [c0x,c0y,c0z, c1x,c1y,c1z, c2x,c2y,c2z, tx,ty,tz]
__device__ __forceinline__ void compose_right(float* __restrict__ P,
                                              float ca, float sa,
                                              float cb, float sb, float Rv) {
  float n0x = ca * P[0] + sa * P[3];
  float n0y = ca * P[1] + sa * P[4];
  float n0z = ca * P[2] + sa * P[5];
  float a1 = -sa * cb, b1 = ca * cb;
  float n1x = a1 * P[0] + b1 * P[3] + sb * P[6];
  float n1y = a1 * P[1] + b1 * P[4] + sb * P[7];
  float n1z = a1 * P[2] + b1 * P[5] + sb * P[8];
  float a2 = sa * sb, b2 = -ca * sb;
  float n2x = a2 * P[0] + b2 * P[3] + cb * P[6];
  float n2y = a2 * P[1] + b2 * P[4] + cb * P[7];
  float n2z = a2 * P[2] + b2 * P[5] + cb * P[8];
  // t' = t + R * newcol0   (since T's translation = R * T's col0)
  P[9]  = fmaf(Rv, n0x, P[9]);
  P[10] = fmaf(Rv, n0y, P[10]);
  P[11] = fmaf(Rv, n0z, P[11]);
  P[0] = n0x; P[1] = n0y; P[2] = n0z;
  P[3] = n1x; P[4] = n1y; P[5] = n1z;
  P[6] = n2x; P[7] = n2y; P[8] = n2z;
}

__launch_bounds__(256)
__global__ void angles2backbone_kernel(const float* __restrict__ angles,
                                       const int* __restrict__ lengths,
                                       float* __restrict__ out) {
  constexpr int L = 512;           // residues per chain
  constexpr int NCHUNK = 256;      // one 6-atom chunk per thread (256*6 = 1536 atoms)
  // 4x4 matrices stored COLUMN-major (16 floats), slot NCHUNK holds identity.
  __shared__ float bufA[(NCHUNK + 1) * 16];
  __shared__ float bufB[(NCHUNK + 1) * 16];

  const int b    = blockIdx.x;
  const int t    = threadIdx.x;    // chunk index 0..255
  const int lane = t & 31;

  // Identity slot in both ping-pong buffers (col-major: offset c*4+r == t).
  if (t < 16) {
    float v = ((t & 3) == (t >> 2)) ? 1.0f : 0.0f;
    bufA[NCHUNK * 16 + t] = v;
    bufB[NCHUNK * 16 + t] = v;
  }

  const float* ang  = angles + (size_t)b * (3 * L);
  const int    nlen = 3 * lengths[b];

  // kappa sin/cos per atom-class (n%3): [C_N_CA, N_CA_C, CA_C_N]
  float cbv[3], sbv[3];
  __sincosf(PI_F - 1.9391f, &sbv[0], &cbv[0]);
  __sincosf(PI_F - 2.0610f, &sbv[1], &cbv[1]);
  __sincosf(PI_F - 2.1186f, &sbv[2], &cbv[2]);
  const float rcv[3] = {1.33f, 1.46f, 1.525f}; // [R_C_N, R_N_CA, R_CA_C]

  // Dihedral sin/cos for this thread's 6 atoms (residues 2t, 2t+1); reused in phase 3.
  const int l0 = 2 * t;
  float cav[6], sav[6];
  {
    float dv[6];
    dv[0] = ang[0 * L + l0];     dv[1] = ang[1 * L + l0];     dv[2] = ang[2 * L + l0];
    dv[3] = ang[0 * L + l0 + 1]; dv[4] = ang[1 * L + l0 + 1]; dv[5] = ang[2 * L + l0 + 1];
#pragma unroll
    for (int e = 0; e < 6; ++e) __sincosf(dv[e], &sav[e], &cav[e]);
  }

  // ---------------- Phase 1: chunk-local product (registers) ----------------
  float P[12] = {1.f, 0.f, 0.f, 0.f, 1.f, 0.f, 0.f, 0.f, 1.f, 0.f, 0.f, 0.f};
#pragma unroll
  for (int e = 0; e < 6; ++e) {
    const int m = e % 3;
    const float Rv = (e == 0 && t == 0) ? 0.0f : rcv[m]; // reference zeroes R[0]
    compose_right(P, cav[e], sav[e], cbv[m], sbv[m], Rv);
  }
  {
    float* w = &bufA[t * 16];                 // column-major 4x4, last row (0,0,0,1)
    w[0]  = P[0]; w[1]  = P[1]; w[2]  = P[2];  w[3]  = 0.f;
    w[4]  = P[3]; w[5]  = P[4]; w[6]  = P[5];  w[7]  = 0.f;
    w[8]  = P[6]; w[9]  = P[7]; w[10] = P[8];  w[11] = 0.f;
    w[12] = P[9]; w[13] = P[10]; w[14] = P[11]; w[15] = 1.f;
  }
  __syncthreads();

  // ------- Phase 2: Hillis-Steele scan of 256 chunk totals via WMMA --------
  // V_WMMA_F32_16X16X4_F32: A(16x4) stacks 4 left 4x4 matrices, B(4x16) holds
  // 4 right 4x4 matrices side-by-side; diagonal 4x4 blocks of C = 4 products.
  float* src = bufA;
  float* dst = bufB;
  const int  waveBase  = (t >> 5) << 5;       // this wave's 32 chunk indices
  const int  Ln        = lane & 15;
  const int  jj        = Ln >> 2;             // block 0..3 within the WMMA
  const int  rA        = Ln & 3;              // A-row within block
  const int  cB        = Ln & 3;              // B-col within block
  const int  klo       = (lane >> 4) << 1;    // K pair: lanes 0-15 -> K0,1; 16-31 -> K2,3
  const bool outActive = (lane < 8) || (lane >= 24);
  const bool hiHalf    = (jj & 1) != 0;       // C VGPRs 4..7 vs 0..3

  for (int step = 1; step < NCHUNK; step <<= 1) {
#pragma unroll
    for (int q = 0; q < 8; ++q) {
      const int tg   = waveBase + (q << 2);   // 4 targets per WMMA
      const int tb   = tg + jj;               // this lane's target chunk
      const int sIdx = (tb >= step) ? (tb - step) : NCHUNK; // identity if out of range
      // A operand: left matrix rows (row-major reads from col-major storage).
      const float* aS = src + sIdx * 16 + rA;
      v2f Av; Av.x = aS[klo * 4]; Av.y = aS[klo * 4 + 4];
      // B operand: right matrix columns -> contiguous float pair.
      const float* bS = src + tb * 16 + cB * 4 + klo;
      v2f Bv; Bv.x = bS[0]; Bv.y = bS[1];
      v8f Cv = {};
      Cv = __builtin_amdgcn_wmma_f32_16x16x4_f32(false, Av, false, Bv,
                                                 (short)0, Cv, false, false);
      // Extract diagonal blocks: 16 active lanes each write one output column.
      if (outActive) {
        float* o = dst + tb * 16 + cB * 4;
        o[0] = hiHalf ? Cv[4] : Cv[0];
        o[1] = hiHalf ? Cv[5] : Cv[1];
        o[2] = hiHalf ? Cv[6] : Cv[2];
        o[3] = hiHalf ? Cv[7] : Cv[3];
      }
    }
    __syncthreads();
    float* tmp = src; src = dst; dst = tmp;   // 8 steps -> ends back in bufA
  }

  // ---------------- Phase 3: apply exclusive prefix, emit coords -----------
  const float* E = src + ((t == 0) ? NCHUNK : (t - 1)) * 16;
  float Q[12];
  Q[0] = E[0];  Q[1]  = E[1];  Q[2]  = E[2];
  Q[3] = E[4];  Q[4]  = E[5];  Q[5]  = E[6];
  Q[6] = E[8];  Q[7]  = E[9];  Q[8]  = E[10];
  Q[9] = E[12]; Q[10] = E[13]; Q[11] = E[14];

  float* ob = out + (size_t)b * (9 * L) + t * 18; // coords (natoms,3) flattened
#pragma unroll
  for (int e = 0; e < 6; ++e) {
    const int m = e % 3;
    const float Rv = (e == 0 && t == 0) ? 0.0f : rcv[m];
    compose_right(Q, cav[e], sav[e], cbv[m], sbv[m], Rv);
    const int  n  = 6 * t + e;
    const bool ok = n < nlen;
    ob[3 * e + 0] = ok ? Q[9]  : 0.0f;
    ob[3 * e + 1] = ok ? Q[10] : 0.0f;
    ob[3 * e + 2] = ok ? Q[11] : 0.0f;
  }
}

extern "C" void kernel_launch(void* const* d_in, const int* in_sizes, int n_in,
                              void* d_out, int out_size, void* d_ws, size_t ws_size,
                              hipStream_t stream) {
  (void)n_in; (void)out_size; (void)d_ws; (void)ws_size;
  const float* angles  = (const float*)d_in[0];
  const int*   lengths = (const int*)d_in[1];
  float*       out     = (float*)d_out;
  const int B = in_sizes[1];   // number of chains (1024)
  angles2backbone_kernel<<<B, 256, 0, stream>>>(angles, lengths, out);
}